// NodeBlock_71425306132748
// MI455X (gfx1250) — compile-verified
//
#include <hip/hip_runtime.h>

typedef __attribute__((ext_vector_type(2))) float v2f;
typedef __attribute__((ext_vector_type(8))) float v8f;

#define N_NODES 50000
#define D 64
#define N_EDGES 800000
#define N_TILES 3125        // 50000 / 16
#define WAVES_PER_BLOCK 8
// LDS layout (floats): [0,49152) W1 fragments, [49152,65536) W2 fragments,
// [65536,67584) per-wave 16x16 hidden staging (8 waves * 256)
#define SMEM_FLOATS (48*16*32*2 + 16*4*4*32*2 + 8*256)

__global__ void zero_ws_kernel(float4* __restrict__ ws, int n4) {
    int i = blockIdx.x * blockDim.x + threadIdx.x;
    if (i < n4) ws[i] = make_float4(0.f, 0.f, 0.f, 0.f);
}

// One thread per (edge, feature): coalesced edge_feat reads, native f32 global atomics.
__global__ void scatter_edges_kernel(const float* __restrict__ ef,
                                     const int* __restrict__ snd,
                                     const int* __restrict__ rcv,
                                     float* __restrict__ agg_in,
                                     float* __restrict__ agg_out) {
    int gid = blockIdx.x * blockDim.x + threadIdx.x;   // < 800000*64 = 51.2M
    int e = gid >> 6;
    int f = gid & 63;
    float v = ef[(size_t)e * 64 + f];
    unsafeAtomicAdd(&agg_in[(size_t)rcv[e] * 64 + f], v);
    unsafeAtomicAdd(&agg_out[(size_t)snd[e] * 64 + f], v);
}

// Fused 2-layer MLP over concat(in_agg, out_agg, node_feat) using f32 WMMA.
// Wave handles one 16-node tile. W1/W2 pre-swizzled in LDS so every B fragment
// is a single aligned 8-byte ds_load.
__global__ __launch_bounds__(256, 1) void node_mlp_kernel(
        const float* __restrict__ nf,
        const float* __restrict__ agg_in,
        const float* __restrict__ agg_out,
        const float* __restrict__ W1,
        const float* __restrict__ b1,
        const float* __restrict__ W2,
        const float* __restrict__ b2,
        float* __restrict__ out) {
    extern __shared__ float smem[];
    const int tid  = threadIdx.x;
    const int lane = tid & 31;
    const int wave = tid >> 5;

    // ---- Stage W1 (192x256) into LDS in WMMA B-fragment order ----
    // fragment id p = (c*48 + kb)*32 + l ; holds {W1[k0][n], W1[k0+1][n]}
    for (int p = tid; p < 48 * 16 * 32; p += 256) {
        int l    = p & 31;
        int kb   = (p >> 5) % 48;
        int c    = (p >> 5) / 48;
        int kofs = (l >> 4) << 1;
        int n    = c * 16 + (l & 15);
        int k0   = kb * 4 + kofs;
        v2f w;
        w.x = W1[(size_t)k0 * 256 + n];
        w.y = W1[(size_t)(k0 + 1) * 256 + n];
        *(v2f*)&smem[(size_t)p * 2] = w;
    }
    // ---- Stage W2 (256x64): fragment id q = ((c*16 + t*4 + kb2)*32 + l) ----
    float* s_w2 = smem + 48 * 16 * 32 * 2;   // offset 49152 floats
    for (int q = tid; q < 16 * 4 * 4 * 32; q += 256) {
        int l    = q & 31;
        int kb2  = (q >> 5) & 3;
        int t    = (q >> 7) & 3;
        int c    = q >> 9;
        int kofs = (l >> 4) << 1;
        int k    = c * 16 + kb2 * 4 + kofs;
        int n    = t * 16 + (l & 15);
        v2f w;
        w.x = W2[(size_t)k * 64 + n];
        w.y = W2[(size_t)(k + 1) * 64 + n];
        *(v2f*)&s_w2[(size_t)q * 2] = w;
    }
    __syncthreads();

    float* s_h = smem + 49152 + 16384 + wave * 256;   // 16x16 hidden staging

    const int tile   = blockIdx.x * WAVES_PER_BLOCK + wave;
    const bool active = (tile < N_TILES);
    const int node0  = (active ? tile : 0) * 16;
    const int row    = lane & 15;           // M for A-fragments / D columns
    const int kofs   = (lane >> 4) << 1;    // K offset within 4-wide K-block
    const int hi     = (lane >> 4) * 8;     // M offset for C/D layout
    const int nloc   = lane & 15;

    // ---- Load all 48 A fragments of X = concat(in_agg, out_agg, node_feat) ----
    v2f afrag[48];
    {
        const float* p0 = agg_in  + (size_t)(node0 + row) * 64 + kofs;
        const float* p1 = agg_out + (size_t)(node0 + row) * 64 + kofs;
        const float* p2 = nf      + (size_t)(node0 + row) * 64 + kofs;
#pragma unroll
        for (int kb = 0; kb < 16; ++kb) {
            afrag[kb]      = *(const v2f*)(p0 + kb * 4);
            afrag[16 + kb] = *(const v2f*)(p1 + kb * 4);
            afrag[32 + kb] = *(const v2f*)(p2 + kb * 4);
        }
    }

    // ---- Output accumulators 16x64, init with b2 broadcast per column ----
    v8f oacc[4];
#pragma unroll
    for (int t = 0; t < 4; ++t) {
        float bv = b2[t * 16 + nloc];
#pragma unroll
        for (int j = 0; j < 8; ++j) oacc[t][j] = bv;
    }

    // ---- Loop over hidden in 16-column chunks: GEMM1 -> ReLU -> GEMM2 ----
    for (int c = 0; c < 16; ++c) {
        float b1v = b1[c * 16 + nloc];
        v8f h;
#pragma unroll
        for (int j = 0; j < 8; ++j) h[j] = b1v;

        const v2f* w1f = (const v2f*)smem + c * 48 * 32 + lane;
#pragma unroll
        for (int kb = 0; kb < 48; ++kb) {
            h = __builtin_amdgcn_wmma_f32_16x16x4_f32(
                    false, afrag[kb], false, w1f[kb * 32],
                    (short)0, h, false, false);
        }

        // ReLU + D-layout -> row-major LDS (wave-private slice)
#pragma unroll
        for (int j = 0; j < 8; ++j) {
            float hv = h[j] > 0.f ? h[j] : 0.f;
            s_h[(hi + j) * 16 + nloc] = hv;
        }
        __syncthreads();

        const v2f* w2f = (const v2f*)s_w2 + c * 16 * 32 + lane;
#pragma unroll
        for (int kb2 = 0; kb2 < 4; ++kb2) {
            v2f a2 = *(const v2f*)&s_h[row * 16 + kb2 * 4 + kofs];
#pragma unroll
            for (int t = 0; t < 4; ++t) {
                oacc[t] = __builtin_amdgcn_wmma_f32_16x16x4_f32(
                        false, a2, false, w2f[(t * 4 + kb2) * 32],
                        (short)0, oacc[t], false, false);
            }
        }
        __syncthreads();
    }

    if (active) {
#pragma unroll
        for (int t = 0; t < 4; ++t)
#pragma unroll
            for (int j = 0; j < 8; ++j)
                out[(size_t)(node0 + hi + j) * 64 + t * 16 + nloc] = oacc[t][j];
    }
}

extern "C" void kernel_launch(void* const* d_in, const int* in_sizes, int n_in,
                              void* d_out, int out_size, void* d_ws, size_t ws_size,
                              hipStream_t stream) {
    const float* node_feat = (const float*)d_in[0];
    const float* edge_feat = (const float*)d_in[1];
    const int*   senders   = (const int*)d_in[2];
    const int*   receivers = (const int*)d_in[3];
    const float* W1        = (const float*)d_in[4];
    const float* b1        = (const float*)d_in[5];
    const float* W2        = (const float*)d_in[6];
    const float* b2        = (const float*)d_in[7];
    float* out = (float*)d_out;

    float* agg_in  = (float*)d_ws;
    float* agg_out = agg_in + (size_t)N_NODES * D;

    // 1) zero the aggregation buffers (must happen every call)
    int n4 = 2 * N_NODES * D / 4;
    zero_ws_kernel<<<(n4 + 255) / 256, 256, 0, stream>>>((float4*)d_ws, n4);

    // 2) edge scatter-add (segment sums by receiver and sender)
    int nthreads = N_EDGES * 64;
    scatter_edges_kernel<<<nthreads / 256, 256, 0, stream>>>(
        edge_feat, senders, receivers, agg_in, agg_out);

    // 3) fused WMMA MLP
    int nblk = (N_TILES + WAVES_PER_BLOCK - 1) / WAVES_PER_BLOCK;  // 391
    node_mlp_kernel<<<nblk, 256, SMEM_FLOATS * sizeof(float), stream>>>(
        node_feat, agg_in, agg_out, W1, b1, W2, b2, out);
}